// StateEncoder_7576322310458
// MI455X (gfx1250) — compile-verified
//
#include <hip/hip_runtime.h>
#include <hip/hip_bf16.h>

#define N_AGENTS 10
#define N_ENEMIES 11
#define N_ENT 21
#define NAA1 7
#define LAS 18
#define EMB 64
#define BSZ 32768

typedef float v2f __attribute__((ext_vector_type(2)));
typedef float v8f __attribute__((ext_vector_type(8)));

#define LDA 66    // padded row stride (floats) for 32x64 e/q/k tiles -> conflict-free ds_load_b64
#define LDSC 34   // padded row stride for 16x32 score^T

struct WaveSmem {
  float e[32 * LDA];
  float q[32 * LDA];
  float k[32 * LDA];
  float energy[32 * 16];
  float scoreT[16 * LDSC];
};

__device__ __forceinline__ v8f wmma4(v2f a, v2f b, v8f c) {
  // D = A(16x4,f32) x B(4x16,f32) + C(16x16,f32), exact f32
  return __builtin_amdgcn_wmma_f32_16x16x4_f32(false, a, false, b, (short)0, c, false, false);
}

__global__ __launch_bounds__(64) void attn_fused_kernel(
    const float* __restrict__ ally,      // (10, BS, 1, 8)
    const float* __restrict__ enemy,     // (11, BS, 1, 6)
    const float* __restrict__ last,      // (10, BS, 1, 18)
    const int*   __restrict__ actions,   // (BS, 10)
    const float* __restrict__ W_ally,    // (64, 22) row-major
    const float* __restrict__ b_ally,    // (64,)
    const float* __restrict__ W_enemy,   // (64, 7)
    const float* __restrict__ b_enemy,   // (64,)
    const float* __restrict__ Wq,        // (64, 64)
    const float* __restrict__ bq,        // (64,)
    const float* __restrict__ Wk,        // (64, 64)
    const float* __restrict__ bk,        // (64,)
    float* __restrict__ out)             // (BS*10, 64) attn_out
{
  __shared__ WaveSmem sm[2];
  const int tid  = threadIdx.x;
  const int wave = tid >> 5;
  const int lane = tid & 31;
  const int lo   = lane & 15;
  const int hi   = lane >> 4;
  const int b    = blockIdx.x * 2 + wave;
  WaveSmem& s = sm[wave];

  // ---- per-batch actions (broadcast loads) ----
  int act[N_AGENTS];
#pragma unroll
  for (int i = 0; i < N_AGENTS; ++i) act[i] = actions[b * N_AGENTS + i];

  // ---- zero padded entity rows 21..31 (incl. pad cols) ----
  for (int idx = lane; idx < 11 * LDA; idx += 32) s.e[21 * LDA + idx] = 0.f;

  // ---- entity embeddings: lane computes output cols d0=lane, d1=lane+32 ----
  {
    const int d0 = lane, d1 = lane + 32;
    float wa0[22], wa1[22];
#pragma unroll
    for (int c = 0; c < 22; ++c) { wa0[c] = W_ally[d0 * 22 + c]; wa1[c] = W_ally[d1 * 22 + c]; }
    const float ba0 = b_ally[d0], ba1 = b_ally[d1];
#pragma unroll
    for (int i = 0; i < N_AGENTS; ++i) {
      const float* af = ally + ((size_t)i * BSZ + b) * 8;
      const float* ls = last + ((size_t)i * BSZ + b) * LAS;
      float vec[22];
#pragma unroll
      for (int c = 0; c < 8; ++c) vec[c] = af[c];
#pragma unroll
      for (int c = 0; c < 7; ++c) vec[8 + c] = (act[i] == c) ? 1.f : 0.f;   // compact_cur
#pragma unroll
      for (int c = 0; c < 7; ++c) vec[15 + c] = ls[c];                      // compact_last
      float a0 = ba0, a1 = ba1;
#pragma unroll
      for (int c = 0; c < 22; ++c) { a0 = fmaf(vec[c], wa0[c], a0); a1 = fmaf(vec[c], wa1[c], a1); }
      s.e[i * LDA + d0] = a0;
      s.e[i * LDA + d1] = a1;
    }
    float we0[7], we1[7];
#pragma unroll
    for (int c = 0; c < 7; ++c) { we0[c] = W_enemy[d0 * 7 + c]; we1[c] = W_enemy[d1 * 7 + c]; }
    const float be0 = b_enemy[d0], be1 = b_enemy[d1];
#pragma unroll
    for (int j = 0; j < N_ENEMIES; ++j) {
      const float* ef = enemy + ((size_t)j * BSZ + b) * 6;
      float vec[7];
#pragma unroll
      for (int c = 0; c < 6; ++c) vec[c] = ef[c];
      bool attacked = false;
#pragma unroll
      for (int i = 0; i < N_AGENTS; ++i) attacked |= (act[i] == NAA1 + j);
      vec[6] = attacked ? 1.f : 0.f;
      float a0 = be0, a1 = be1;
#pragma unroll
      for (int c = 0; c < 7; ++c) { a0 = fmaf(vec[c], we0[c], a0); a1 = fmaf(vec[c], we1[c], a1); }
      s.e[(N_AGENTS + j) * LDA + d0] = a0;
      s.e[(N_AGENTS + j) * LDA + d1] = a1;
    }
  }
  __syncthreads();

  // ---- q = e@Wq^T + bq ; k = e@Wk^T + bk  (32x64, WMMA f32 16x16x4) ----
#pragma unroll
  for (int qk = 0; qk < 2; ++qk) {
    const float* W    = qk ? Wk : Wq;
    const float* bias = qk ? bk : bq;
    float* dst        = qk ? s.k : s.q;
#pragma unroll
    for (int mt = 0; mt < 2; ++mt) {
#pragma unroll
      for (int nt = 0; nt < 4; ++nt) {
        const int n = nt * 16 + lo;
        const float bv = bias[n];
        v8f acc = {bv, bv, bv, bv, bv, bv, bv, bv};
#pragma unroll
        for (int ks = 0; ks < 16; ++ks) {
          const int kb = ks * 4 + hi * 2;
          v2f a, bb;
          a.x  = s.e[(mt * 16 + lo) * LDA + kb];
          a.y  = s.e[(mt * 16 + lo) * LDA + kb + 1];
          bb.x = W[n * 64 + kb];         // B[a,n] = W[n,a] (row-contiguous)
          bb.y = W[n * 64 + kb + 1];
          acc = wmma4(a, bb, acc);
        }
#pragma unroll
        for (int v = 0; v < 8; ++v)
          dst[(mt * 16 + hi * 8 + v) * LDA + n] = acc[v];
      }
    }
  }
  __syncthreads();

  // ---- energy[q,ki] = (q . k)/8 ; only ki tile 0 (cols 0..15) is needed ----
#pragma unroll
  for (int mt = 0; mt < 2; ++mt) {
    v8f acc = {};
#pragma unroll
    for (int ks = 0; ks < 16; ++ks) {
      const int kb = ks * 4 + hi * 2;
      v2f a, bb;
      a.x  = s.q[(mt * 16 + lo) * LDA + kb];
      a.y  = s.q[(mt * 16 + lo) * LDA + kb + 1];
      bb.x = s.k[lo * LDA + kb];         // B[a,n] = k[n,a] (row-contiguous)
      bb.y = s.k[lo * LDA + kb + 1];
      acc = wmma4(a, bb, acc);
    }
#pragma unroll
    for (int v = 0; v < 8; ++v)
      s.energy[(mt * 16 + hi * 8 + v) * 16 + lo] = acc[v] * 0.125f;
  }
  __syncthreads();

  // ---- column softmax over q = 0..20 (mask padded rows), write score^T (16x32) ----
  for (int idx = lane; idx < 16 * LDSC; idx += 32) s.scoreT[idx] = 0.f;
  __syncthreads();
  if (lane < N_AGENTS) {
    const int col = lane;
    float mx = -1e30f;
#pragma unroll
    for (int m = 0; m < N_ENT; ++m) mx = fmaxf(mx, s.energy[m * 16 + col]);
    float ssum = 0.f;
    float ex[N_ENT];
#pragma unroll
    for (int m = 0; m < N_ENT; ++m) { ex[m] = __expf(s.energy[m * 16 + col] - mx); ssum += ex[m]; }
    const float inv = 1.f / ssum;
#pragma unroll
    for (int m = 0; m < N_ENT; ++m) s.scoreT[col * LDSC + m] = ex[m] * inv;
  }
  __syncthreads();

  // ---- attn_out[ki,d] = sum_q scoreT[ki,q] * e[q,d]; ki = 0..9 ----
#pragma unroll
  for (int nt = 0; nt < 4; ++nt) {
    v8f acc = {};
#pragma unroll
    for (int ks = 0; ks < 8; ++ks) {
      const int kb = ks * 4 + hi * 2;
      v2f a, bb;
      a.x  = s.scoreT[lo * LDSC + kb];
      a.y  = s.scoreT[lo * LDSC + kb + 1];
      bb.x = s.e[kb * LDA + nt * 16 + lo];        // B[a,n] = e[q,d]
      bb.y = s.e[(kb + 1) * LDA + nt * 16 + lo];
      acc = wmma4(a, bb, acc);
    }
    const int d = nt * 16 + lo;
#pragma unroll
    for (int v = 0; v < 8; ++v) {
      const int ki = hi * 8 + v;
      if (ki < N_AGENTS)
        out[((size_t)b * N_AGENTS + ki) * 64 + d] = acc[v];
    }
  }
}

extern "C" void kernel_launch(void* const* d_in, const int* in_sizes, int n_in,
                              void* d_out, int out_size, void* d_ws, size_t ws_size,
                              hipStream_t stream) {
  (void)in_sizes; (void)n_in; (void)d_ws; (void)ws_size; (void)out_size;
  const float* ally   = (const float*)d_in[0];
  const float* enemy  = (const float*)d_in[1];
  const float* last   = (const float*)d_in[2];
  const float* hidden = (const float*)d_in[3];
  const int*   acts   = (const int*)d_in[4];
  const float* W_ally = (const float*)d_in[5];
  const float* b_ally = (const float*)d_in[6];
  const float* W_en   = (const float*)d_in[7];
  const float* b_en   = (const float*)d_in[8];
  const float* Wq     = (const float*)d_in[9];
  const float* bq     = (const float*)d_in[10];
  const float* Wk     = (const float*)d_in[11];
  const float* bk     = (const float*)d_in[12];
  float* out = (float*)d_out;

  attn_fused_kernel<<<BSZ / 2, 64, 0, stream>>>(ally, enemy, last, acts,
                                                W_ally, b_ally, W_en, b_en,
                                                Wq, bq, Wk, bk, out);

  // hidden_state pass-through: second output, flat-concatenated after attn_out
  const size_t hid = (size_t)BSZ * N_AGENTS * EMB;
  hipMemcpyAsync(out + hid, hidden, hid * sizeof(float),
                 hipMemcpyDeviceToDevice, stream);
}